// Model_6932077216150
// MI455X (gfx1250) — compile-verified
//
#include <hip/hip_runtime.h>
#include <stdint.h>

// BRITS data prep for MI455X (gfx1250, wave32).
// out[0] = forward deltas, out[1] = backward deltas (abs), out[2] = flip(X, axis=s).
// Bandwidth-bound (~1.2 GB traffic -> ~52us floor @ 23.3 TB/s). No matrix math in
// this op => no WMMA; CDNA5 paths used: global_load_async_to_lds_b32 (+ s_wait_asynccnt),
// global_prefetch_b8 deep software pipeline, nontemporal TH hints, LDS broadcast.

typedef float v4f __attribute__((ext_vector_type(4)));
typedef int   v4i __attribute__((ext_vector_type(4)));

constexpr int B = 32, S = 2048, F = 768;
constexpr int TPB          = 192;      // F/4 float4-chains per (b,dir) row; 6 wave32s
constexpr int DELTA_BLOCKS = 2 * B;    // {fwd,bwd} x batch
constexpr int COPY_BLOCKS  = S;        // one fixed time-row per copy block (no div/mod)
constexpr int PF_DIST      = 32;       // prefetch distance in time rows (96KB/block ahead)

// One thread walks the 2048-step recurrence for 4 adjacent features.
// mask row index at step t is (START + DIR*t): fwd START=-1,DIR=+1 ; bwd START=S,DIR=-1.
// Backward uses positive reversed gaps => result equals abs(reference) exactly.
template <int DIR, int START>
__device__ __forceinline__ void delta_chain(const int* __restrict__ mbase,
                                            float* __restrict__ obase,
                                            const float* __restrict__ gap_sh)
{
    float d0 = 0.f, d1 = 0.f, d2 = 0.f, d3 = 0.f;
    v4f z = {0.f, 0.f, 0.f, 0.f};
    __builtin_nontemporal_store(z, (v4f*)obase);     // delta row 0 is zero

    int t = 1;
    // Main loop: unconditional global_prefetch_b8 PF_DIST rows ahead.
    #pragma unroll 8
    for (; t <= S - 1 - PF_DIST; ++t) {
        const long mrow = (long)START + (long)DIR * (long)t;
        v4i m = __builtin_nontemporal_load((const v4i*)(mbase + mrow * (long)F));
        const long prow = mrow + (long)(DIR * PF_DIST);
        __builtin_prefetch((const void*)(mbase + prow * (long)F), 0, 3);

        const float g = gap_sh[t];                   // LDS broadcast (uniform addr)
        d0 = fmaf((float)(1 - m.x), d0, g);
        d1 = fmaf((float)(1 - m.y), d1, g);
        d2 = fmaf((float)(1 - m.z), d2, g);
        d3 = fmaf((float)(1 - m.w), d3, g);
        v4f dv = {d0, d1, d2, d3};
        __builtin_nontemporal_store(dv, (v4f*)(obase + (long)t * F));
    }
    // Tail: last PF_DIST rows were already prefetched; no guard needed.
    #pragma unroll 8
    for (; t < S; ++t) {
        const long mrow = (long)START + (long)DIR * (long)t;
        v4i m = __builtin_nontemporal_load((const v4i*)(mbase + mrow * (long)F));
        const float g = gap_sh[t];
        d0 = fmaf((float)(1 - m.x), d0, g);
        d1 = fmaf((float)(1 - m.y), d1, g);
        d2 = fmaf((float)(1 - m.z), d2, g);
        d3 = fmaf((float)(1 - m.w), d3, g);
        v4f dv = {d0, d1, d2, d3};
        __builtin_nontemporal_store(dv, (v4f*)(obase + (long)t * F));
    }
}

__global__ __launch_bounds__(TPB)
void brits_prepare_kernel(const float* __restrict__ ts,
                          const float* __restrict__ X,
                          const int*   __restrict__ masks,
                          float*       __restrict__ out)
{
    __shared__ float ts_sh[S];                       // raw timestamps (async-DMA'd)
    __shared__ float gap_sh[S];                      // per-direction gaps
    const size_t SEC = (size_t)B * S * F;            // one output component

    if (blockIdx.x < (unsigned)DELTA_BLOCKS) {
        // ---------------- recurrence blocks (scheduled first: long pole) ----------
        const int b   = (int)(blockIdx.x >> 1);
        const int bwd = (int)(blockIdx.x & 1);
        const float* tsb = ts + (size_t)b * S;

        // CDNA5 async global->LDS DMA of the 8KB timestamp row (ASYNCcnt-tracked,
        // bypasses VGPRs). Low 32 bits of a generic LDS pointer = LDS byte offset.
        for (int i = threadIdx.x; i < S; i += TPB) {
            uint32_t lds_off = (uint32_t)(uintptr_t)&ts_sh[i];
            uint32_t voff    = (uint32_t)i * 4u;
            asm volatile("global_load_async_to_lds_b32 %0, %1, %2"
                         :: "v"(lds_off), "v"(voff), "s"(tsb)
                         : "memory");
        }
        asm volatile("s_wait_asynccnt 0" ::: "memory");
        __syncthreads();

        // Gaps from LDS (backward uses positive reversed gaps == exact abs()).
        for (int i = threadIdx.x; i < S; i += TPB) {
            float g = 0.f;
            if (i > 0)
                g = bwd ? (ts_sh[S - i] - ts_sh[S - 1 - i])
                        : (ts_sh[i]     - ts_sh[i - 1]);
            gap_sh[i] = g;
        }
        __syncthreads();

        const size_t base  = (size_t)b * S * F + (size_t)threadIdx.x * 4;
        const int*  mbase  = masks + base;
        float*      obase  = out + (size_t)bwd * SEC + base;

        if (bwd) delta_chain<-1, S>(mbase, obase, gap_sh);
        else     delta_chain<+1, -1>(mbase, obase, gap_sh);
    } else {
        // ---------------- flip-copy: out[2][b][t][f] = X[b][S-1-t][f] ------------
        // Block owns a fixed t; iterates over b with constant S*F stride.
        const int t = (int)(blockIdx.x - DELTA_BLOCKS);          // 0..S-1
        const float* src = X + (size_t)(S - 1 - t) * F + (size_t)threadIdx.x * 4;
        float*       dst = out + 2 * SEC + (size_t)t * F + (size_t)threadIdx.x * 4;

        #pragma unroll 4
        for (int b = 0; b < B; ++b) {
            const size_t off = (size_t)b * S * F;
            v4f v = __builtin_nontemporal_load((const v4f*)(src + off));
            __builtin_nontemporal_store(v, (v4f*)(dst + off));
        }
    }
}

extern "C" void kernel_launch(void* const* d_in, const int* in_sizes, int n_in,
                              void* d_out, int out_size, void* d_ws, size_t ws_size,
                              hipStream_t stream)
{
    (void)in_sizes; (void)n_in; (void)out_size; (void)d_ws; (void)ws_size;
    const float* ts    = (const float*)d_in[0];   // timestamps   [B,S]   f32
    const float* X     = (const float*)d_in[1];   // encoded_data [B,S,F] f32
    const int*   masks = (const int*)  d_in[2];   // masks        [B,S,F] i32
    float*       out   = (float*)d_out;           // [3,B,S,F] f32

    brits_prepare_kernel<<<dim3(DELTA_BLOCKS + COPY_BLOCKS), dim3(TPB), 0, stream>>>(
        ts, X, masks, out);
}